// VectorQuantizer_27152783245818
// MI455X (gfx1250) — compile-verified
//
#include <hip/hip_runtime.h>
#include <hip/hip_bf16.h>

// ---------------------------------------------------------------------------
// VQ quantizer forward for MI455X (gfx1250, wave32).
//   dist = ||z||^2 - 2 z.E^T + ||E||^2 ; argmin ; gather ; loss
// GEMM part done with V_WMMA_F32_16X16X4_F32 (exact fp32) fused with argmin.
// ---------------------------------------------------------------------------

typedef float v2f __attribute__((ext_vector_type(2)));
typedef float v8f __attribute__((ext_vector_type(8)));

#define N_E      1024
#define E_DIM    256
#define N_ROWS   65536          // 16*4096
#define ROWS_PB  128            // rows per block (8 waves * 16)
#define NBLK     (N_ROWS / ROWS_PB)   // 512
#define LSTRIDE  260            // padded LDS row stride (floats) -> bank-conflict-free
#define ZQ_ELEMS (N_ROWS * E_DIM)     // 16777216

// ---------------- kernel 0: codebook norms ||E_j||^2 -> ws ----------------
__global__ void vq_cbnorm_kernel(const float* __restrict__ cb,
                                 float* __restrict__ cbn) {
    int code = blockIdx.x * blockDim.x + threadIdx.x;   // 0..1023
    const float4* c4 = (const float4*)cb + code * (E_DIM / 4);
    float s = 0.f;
#pragma unroll
    for (int j = 0; j < E_DIM / 4; ++j) {
        float4 v = c4[j];
        s += v.x * v.x + v.y * v.y + v.z * v.z + v.w * v.w;
    }
    cbn[code] = s;
}

// ---------------- kernel 1: fused distance GEMM + argmin + gather ----------
__global__ void __launch_bounds__(256)
vq_main_kernel(const float* __restrict__ z,
               const float* __restrict__ cb,
               const float* __restrict__ cbnG,
               float* __restrict__ out_zq,
               float* __restrict__ out_idx,
               float* __restrict__ ws_partials) {
    extern __shared__ float sm[];
    float* zl  = sm;                       // 128*260 = 33280 floats
    float* cbt = zl + ROWS_PB * LSTRIDE;   // 16*260  =  4160 floats
    float* cbn = cbt + 16 * LSTRIDE;       // 1024 floats
    float* red = cbn + N_E;                // 256 floats
    int*  sidx = (int*)(red + 256);        // 128 ints

    const int tid  = threadIdx.x;
    const int wave = tid >> 5;
    const int lane = tid & 31;
    const int col  = lane & 15;            // code column / row-in-tile selector
    const int hi   = lane >> 4;            // half-wave: K-pair / M-offset selector
    const int row0 = blockIdx.x * ROWS_PB;

    const float4* z4  = (const float4*)z;
    const float4* cb4 = (const float4*)cb;

    // stage z tile (128 rows x 256 f32) into padded LDS
    for (int u = tid; u < ROWS_PB * (E_DIM / 4); u += 256) {
        int r = u >> 6, c = u & 63;
        *(float4*)(zl + r * LSTRIDE + c * 4) = z4[(size_t)(row0 + r) * (E_DIM / 4) + c];
    }
    // stage codebook norms
    for (int u = tid; u < N_E; u += 256) cbn[u] = cbnG[u];

    // per-lane running minima: VGPR r of the C/D tile holds row (r + 8*hi)
    float minv[8];
    int   mini[8];
#pragma unroll
    for (int r = 0; r < 8; ++r) { minv[r] = 3.4e38f; mini[r] = 0; }

    // A-fragment base: lane reads z[wave*16 + col][k + 2*hi .. +1]
    const float* za = zl + (wave * 16 + col) * LSTRIDE + 2 * hi;

    for (int t = 0; t < 64; ++t) {          // 64 tiles of 16 codes
        const int c0 = t * 16;
        __syncthreads();
        // cooperative load of 16 codebook rows into padded LDS
        for (int u = tid; u < 16 * (E_DIM / 4); u += 256) {
            int cd = u >> 6, c = u & 63;
            *(float4*)(cbt + cd * LSTRIDE + c * 4) = cb4[(size_t)(c0 + cd) * (E_DIM / 4) + c];
        }
        __syncthreads();
        if (t < 63) {   // prefetch next tile's lines into cache hierarchy
            for (int u = tid; u < 16 * (E_DIM / 4); u += 256) {
                int cd = u >> 6, c = u & 63;
                __builtin_prefetch((const void*)&cb4[(size_t)(c0 + 16 + cd) * (E_DIM / 4) + c], 0, 1);
            }
        }

        const float* ba = cbt + col * LSTRIDE + 2 * hi;  // B-fragment base

        v8f a0 = {}, a1 = {}, a2 = {}, a3 = {};          // 4 accs break RAW chain
#pragma unroll
        for (int k = 0; k < E_DIM; k += 16) {
            v2f va0 = *(const v2f*)(za + k);
            v2f vb0 = *(const v2f*)(ba + k);
            a0 = __builtin_amdgcn_wmma_f32_16x16x4_f32(false, va0, false, vb0,
                                                       (short)0, a0, false, false);
            v2f va1 = *(const v2f*)(za + k + 4);
            v2f vb1 = *(const v2f*)(ba + k + 4);
            a1 = __builtin_amdgcn_wmma_f32_16x16x4_f32(false, va1, false, vb1,
                                                       (short)0, a1, false, false);
            v2f va2 = *(const v2f*)(za + k + 8);
            v2f vb2 = *(const v2f*)(ba + k + 8);
            a2 = __builtin_amdgcn_wmma_f32_16x16x4_f32(false, va2, false, vb2,
                                                       (short)0, a2, false, false);
            v2f va3 = *(const v2f*)(za + k + 12);
            v2f vb3 = *(const v2f*)(ba + k + 12);
            a3 = __builtin_amdgcn_wmma_f32_16x16x4_f32(false, va3, false, vb3,
                                                       (short)0, a3, false, false);
        }
        v8f d = (a0 + a1) + (a2 + a3);      // dot(z_row, code)

        // score = ||E||^2 - 2*dot  (||z||^2 constant per row -> irrelevant)
        const float cn = cbn[c0 + col];
#pragma unroll
        for (int r = 0; r < 8; ++r) {
            float s = cn - 2.0f * d[r];
            bool lt = s < minv[r];          // strict < keeps earliest code index
            mini[r] = lt ? (c0 + col) : mini[r];
            minv[r] = lt ? s : minv[r];
        }
    }

    // cross-lane argmin within each 16-lane half (lowest index wins ties)
#pragma unroll
    for (int r = 0; r < 8; ++r) {
        float v = minv[r];
        int   i = mini[r];
#pragma unroll
        for (int off = 8; off >= 1; off >>= 1) {
            float ov = __shfl_xor(v, off, 32);
            int   oi = __shfl_xor(i, off, 32);
            if (ov < v || (ov == v && oi < i)) { v = ov; i = oi; }
        }
        if (col == 0) {
            int rl = wave * 16 + r + 8 * hi;
            sidx[rl] = i;
            out_idx[row0 + rl] = (float)i;
        }
    }
    __syncthreads();

    // phase 2: gather z_q = codebook[idx], write out, accumulate loss partial
    float lsum = 0.f;
    float4* oz4 = (float4*)out_zq;
    for (int u = tid; u < ROWS_PB * (E_DIM / 4); u += 256) {
        int r = u >> 6, c = u & 63;
        int code = sidx[r];
        float4 q  = cb4[(size_t)code * (E_DIM / 4) + c];
        float4 zv = *(const float4*)(zl + r * LSTRIDE + c * 4);
        float dx = q.x - zv.x, dy = q.y - zv.y, dz = q.z - zv.z, dw = q.w - zv.w;
        lsum += dx * dx + dy * dy + dz * dz + dw * dw;
        oz4[(size_t)(row0 + r) * (E_DIM / 4) + c] = q;
    }
    red[tid] = lsum;
    __syncthreads();
#pragma unroll
    for (int s = 128; s > 0; s >>= 1) {     // fixed-order tree -> deterministic
        if (tid < s) red[tid] += red[tid + s];
        __syncthreads();
    }
    if (tid == 0) ws_partials[blockIdx.x] = red[0];
}

// ---------------- kernel 2: deterministic final loss reduction -------------
__global__ void vq_finalize_kernel(const float* __restrict__ ws_partials,
                                   float* __restrict__ out_loss) {
    __shared__ float red[256];
    int tid = threadIdx.x;
    red[tid] = ws_partials[tid] + ws_partials[tid + 256];
    __syncthreads();
#pragma unroll
    for (int s = 128; s > 0; s >>= 1) {
        if (tid < s) red[tid] += red[tid + s];
        __syncthreads();
    }
    if (tid == 0) {
        // loss = mean((zq-z)^2) + BETA*mean((z-zq)^2) = 1.25 * sum / numel
        out_loss[0] = red[0] * 1.25f / (float)ZQ_ELEMS;
    }
}

extern "C" void kernel_launch(void* const* d_in, const int* in_sizes, int n_in,
                              void* d_out, int out_size, void* d_ws, size_t ws_size,
                              hipStream_t stream) {
    const float* z  = (const float*)d_in[0];   // [16,4096,256] f32
    const float* cb = (const float*)d_in[1];   // [1024,256] f32

    float* out_zq   = (float*)d_out;           // 16777216 floats
    float* out_loss = out_zq + ZQ_ELEMS;       // 1 float
    float* out_idx  = out_zq + ZQ_ELEMS + 1;   // 65536 floats (indices as f32)

    float* ws       = (float*)d_ws;            // [0..511] loss partials
    float* ws_cbn   = ws + NBLK;               // [512..1535] codebook norms

    vq_cbnorm_kernel<<<N_E / 256, 256, 0, stream>>>(cb, ws_cbn);

    size_t smem = (size_t)(ROWS_PB * LSTRIDE    // z tile
                           + 16 * LSTRIDE       // codebook tile
                           + N_E                // codebook norms
                           + 256) * sizeof(float)
                  + ROWS_PB * sizeof(int);      // argmin indices
    vq_main_kernel<<<NBLK, 256, smem, stream>>>(z, cb, ws_cbn,
                                                out_zq, out_idx, ws);

    vq_finalize_kernel<<<1, 256, 0, stream>>>(ws, out_loss);
}